// ReachabilityGNN_3126736191959
// MI455X (gfx1250) — compile-verified
//
#include <hip/hip_runtime.h>

// ---------------------------------------------------------------------------
// GCN (2x GCNConv + linear head) for gfx1250.
// Dense transforms: V_WMMA_F32_16X16X4_F32, fully unrolled, branch-free loads.
// Sparse aggregation: L2-resident gather + f32 atomic scatter; edge stream
// uses non-temporal loads so it doesn't evict the resident feature table.
// ---------------------------------------------------------------------------

typedef __attribute__((ext_vector_type(2))) float v2f;
typedef __attribute__((ext_vector_type(8))) float v8f;

#define N_NODES 100000
#define N_EDGES 6400000
#define NODE_IN 10
#define HIDDEN  32
#define OUT_DIM 4

typedef long long edge_t;   // reference builds edge_index with dtype=jnp.int64

// ---------------- init: deg = 1.0 (self loop), agg = 0 ----------------------
__global__ void k_init(float* __restrict__ deg, float* __restrict__ agg,
                       int n, int aggn) {
  int i = blockIdx.x * blockDim.x + threadIdx.x;
  int stride = gridDim.x * blockDim.x;
  for (int j = i; j < n; j += stride)    deg[j] = 1.0f;
  for (int j = i; j < aggn; j += stride) agg[j] = 0.0f;
}

__global__ void k_zero(float* __restrict__ p, int n) {
  int i = blockIdx.x * blockDim.x + threadIdx.x;
  int stride = gridDim.x * blockDim.x;
  for (int j = i; j < n; j += stride) p[j] = 0.0f;
}

// ---------------- degree: deg[dst] += 1 -------------------------------------
__global__ void k_degree(const edge_t* __restrict__ dst, float* __restrict__ deg) {
  int i = blockIdx.x * blockDim.x + threadIdx.x;
  int stride = gridDim.x * blockDim.x;
  for (int e = i; e < N_EDGES; e += stride) {
    __builtin_prefetch(dst + e + 4096, 0, 1);          // global_prefetch_b8
    const int d = (int)__builtin_nontemporal_load(dst + e);   // TH=NT stream
    atomicAdd(&deg[d], 1.0f);
  }
}

// ---------------- deg -> dinv in place --------------------------------------
__global__ void k_dinv(float* __restrict__ deg) {
  int i = blockIdx.x * blockDim.x + threadIdx.x;
  if (i < N_NODES) {
    float d = deg[i];
    deg[i] = (d > 0.0f) ? rsqrtf(d) : 0.0f;
  }
}

// ---------------- WMMA GEMM: out[M x NCOLS] = in[M x K] @ W[K x LDW] (+bias)
// One wave computes one 16x16 output tile via V_WMMA_F32_16X16X4_F32.
// Fragment layouts per CDNA5 ISA 7.12.2:
//   A (16x4 f32, 2 VGPR): lane l(0-15)/half h: a[v] = A[M=l][K = 2h + v]
//   B (4x16 f32, 2 VGPR): b[v] = B[K = 2h + v][N = l]
//   C/D (16x16 f32, 8 VGPR): d[r] = D[M = r + 8h][N = l]
// All loads are unconditional + in-range; padding handled by 0/1 multipliers
// (no exec-mask branches inside the WMMA loop).
// ---------------------------------------------------------------------------
template <int K, int LDIN, int LDW, int NCOLS, int TILES_N, int LDOUT, bool BIAS>
__global__ void k_gemm_wmma(const float* __restrict__ in,
                            const float* __restrict__ W,
                            const float* __restrict__ bias,
                            float* __restrict__ out) {
  constexpr int TILES_M = N_NODES / 16;              // 6250, exact
  const int wave = (blockIdx.x * blockDim.x + threadIdx.x) >> 5;
  const int lane = threadIdx.x & 31;
  if (wave >= TILES_M * TILES_N) return;             // wave-uniform branch

  const int tm = wave / TILES_N;
  const int tn = wave - tm * TILES_N;
  const int m0 = tm * 16;
  const int n0 = tn * 16;
  const int half = lane >> 4;
  const int l    = lane & 15;
  const int col  = n0 + l;
  const int arow = m0 + l;

  constexpr bool FULLCOLS = (NCOLS % 16 == 0);
  const int   colc  = FULLCOLS ? col : (col < NCOLS ? col : NCOLS - 1);
  const float cmask = FULLCOLS ? 1.0f : (col < NCOLS ? 1.0f : 0.0f);

  v8f acc = {};
  constexpr int KFULL = (K / 4) * 4;

#pragma unroll
  for (int kc = 0; kc < KFULL; kc += 4) {
    const int k0 = kc + 2 * half;                    // always < K here
    const float2 av = *(const float2*)(in + (size_t)arow * LDIN + k0);
    v2f a, b;
    a[0] = av.x;
    a[1] = av.y;
    b[0] = W[(size_t)k0 * LDW + colc] * cmask;
    b[1] = W[(size_t)(k0 + 1) * LDW + colc] * cmask;
    acc = __builtin_amdgcn_wmma_f32_16x16x4_f32(false, a, false, b,
                                                (short)0, acc, false, false);
  }

  if constexpr (KFULL < K) {
    // Remainder: K % 4 == 2 (K=10). half=0 lanes carry k=KFULL,KFULL+1;
    // half=1 lanes would carry k=KFULL+2,3 (out of range) -> masked to zero.
    const float hm = (half == 0) ? 1.0f : 0.0f;
    const float2 av = *(const float2*)(in + (size_t)arow * LDIN + KFULL);
    v2f a, b;
    a[0] = av.x * hm;
    a[1] = av.y * hm;
    b[0] = W[(size_t)KFULL * LDW + colc] * hm * cmask;
    b[1] = W[(size_t)(KFULL + 1) * LDW + colc] * hm * cmask;
    acc = __builtin_amdgcn_wmma_f32_16x16x4_f32(false, a, false, b,
                                                (short)0, acc, false, false);
  }

  if (FULLCOLS || col < NCOLS) {
    float bb;
    if constexpr (BIAS) bb = bias[colc];
#pragma unroll
    for (int r = 0; r < 8; ++r) {
      float v = acc[r];
      if constexpr (BIAS) v += bb;
      out[(size_t)(m0 + r + 8 * half) * LDOUT + col] = v;
    }
  }
}

// ---------------- edge scatter: agg[dst] += h[src] * dinv[src]*dinv[dst] ----
// 4 threads per edge, 8 floats each (two float4 gathers + 8 f32 atomics).
// Consecutive lanes share the same edge cacheline -> edge list read ~once.
// Edge stream is non-temporal; feature table + accumulator stay L2-resident.
#define TPE 4
__global__ void k_scatter(const edge_t* __restrict__ ei,
                          const float* __restrict__ dinv,
                          const float* __restrict__ h,
                          float* __restrict__ agg) {
  long long idx = (long long)blockIdx.x * blockDim.x + threadIdx.x;
  const long long total = (long long)N_EDGES * TPE;
  const long long stride = (long long)gridDim.x * blockDim.x;
  for (long long t = idx; t < total; t += stride) {
    const int e = (int)(t >> 2);
    const int c = (int)(t & 3);           // which 8-float chunk of 32
    __builtin_prefetch(ei + e + 2048, 0, 1);             // src stream
    __builtin_prefetch(ei + N_EDGES + e + 2048, 0, 1);   // dst stream
    const int s = (int)__builtin_nontemporal_load(ei + e);
    const int d = (int)__builtin_nontemporal_load(ei + N_EDGES + e);
    const float norm = dinv[s] * dinv[d];
    const float4* hs = (const float4*)(h + (size_t)s * HIDDEN + c * 8);
    float*       ad = agg + (size_t)d * HIDDEN + c * 8;
    const float4 v0 = hs[0];
    const float4 v1 = hs[1];
    atomicAdd(ad + 0, v0.x * norm);
    atomicAdd(ad + 1, v0.y * norm);
    atomicAdd(ad + 2, v0.z * norm);
    atomicAdd(ad + 3, v0.w * norm);
    atomicAdd(ad + 4, v1.x * norm);
    atomicAdd(ad + 5, v1.y * norm);
    atomicAdd(ad + 6, v1.z * norm);
    atomicAdd(ad + 7, v1.w * norm);
  }
}

// ---------------- epilogue: agg = relu(agg + h*dinv^2 + bias) ---------------
__global__ void k_epilogue(const float* __restrict__ h,
                           const float* __restrict__ dinv,
                           const float* __restrict__ bias,
                           float* __restrict__ agg) {
  int t = blockIdx.x * blockDim.x + threadIdx.x;   // N_NODES * 8 float4-chunks
  if (t >= N_NODES * 8) return;
  const int node = t >> 3;
  const int f0   = (t & 7) * 4;
  const float di = dinv[node];
  const float sl = di * di;                        // self-loop norm
  float* ap = agg + (size_t)node * HIDDEN + f0;
  const float* hp = h + (size_t)node * HIDDEN + f0;
  float4 av = *(const float4*)ap;
  const float4 hv = *(const float4*)hp;
  av.x = fmaxf(av.x + hv.x * sl + bias[f0 + 0], 0.0f);
  av.y = fmaxf(av.y + hv.y * sl + bias[f0 + 1], 0.0f);
  av.z = fmaxf(av.z + hv.z * sl + bias[f0 + 2], 0.0f);
  av.w = fmaxf(av.w + hv.w * sl + bias[f0 + 3], 0.0f);
  *(float4*)ap = av;
}

// ---------------------------------------------------------------------------
extern "C" void kernel_launch(void* const* d_in, const int* in_sizes, int n_in,
                              void* d_out, int out_size, void* d_ws, size_t ws_size,
                              hipStream_t stream) {
  (void)in_sizes; (void)n_in; (void)out_size; (void)ws_size;

  const float*  x   = (const float*)d_in[0];
  const edge_t* ei  = (const edge_t*)d_in[1];   // [2, E] row-major, int64
  const float*  W1  = (const float*)d_in[3];
  const float*  b1  = (const float*)d_in[4];
  const float*  W2  = (const float*)d_in[5];
  const float*  b2  = (const float*)d_in[6];
  const float*  Wc  = (const float*)d_in[7];
  const float*  bc  = (const float*)d_in[8];
  float*        out = (float*)d_out;

  float* ws   = (float*)d_ws;
  float* dinv = ws;                               // N
  float* A    = ws + N_NODES;                     // N x 32 (transformed feats)
  float* B    = A + (size_t)N_NODES * HIDDEN;     // N x 32 (aggregation acc)

  const int featN = N_NODES * HIDDEN;

  // 1) deg = 1 (self-loop), B = 0
  k_init<<<(featN + 255) / 256, 256, 0, stream>>>(dinv, B, N_NODES, featN);
  // 2) deg[dst] += 1 per edge
  k_degree<<<(N_EDGES + 255) / 256, 256, 0, stream>>>(ei + N_EDGES, dinv);
  // 3) dinv = rsqrt(deg)
  k_dinv<<<(N_NODES + 255) / 256, 256, 0, stream>>>(dinv);

  const int gemmBlocks2 = ((N_NODES / 16) * 2 + 3) / 4;  // 2 N-tiles, 4 waves/blk
  const int gemmBlocks1 = ((N_NODES / 16) * 1 + 3) / 4;  // 1 N-tile
  const long long scatterItems = (long long)N_EDGES * TPE;
  const int scatterBlocks = (int)((scatterItems + 255) / 256);
  const int epiBlocks = (N_NODES * 8 + 255) / 256;

  // ----- Layer 1: A = x @ W1 -----
  k_gemm_wmma<NODE_IN, NODE_IN, HIDDEN, HIDDEN, 2, HIDDEN, false>
      <<<gemmBlocks2, 128, 0, stream>>>(x, W1, nullptr, A);
  k_scatter<<<scatterBlocks, 256, 0, stream>>>(ei, dinv, A, B);
  k_epilogue<<<epiBlocks, 256, 0, stream>>>(A, dinv, b1, B);   // B = H1

  // ----- Layer 2: A = H1 @ W2 -----
  k_gemm_wmma<HIDDEN, HIDDEN, HIDDEN, HIDDEN, 2, HIDDEN, false>
      <<<gemmBlocks2, 128, 0, stream>>>(B, W2, nullptr, A);
  k_zero<<<(featN + 255) / 256, 256, 0, stream>>>(B, featN);
  k_scatter<<<scatterBlocks, 256, 0, stream>>>(ei, dinv, A, B);
  k_epilogue<<<epiBlocks, 256, 0, stream>>>(A, dinv, b2, B);   // B = H2

  // ----- Classifier: out = H2 @ Wc + bc (4 valid cols of a 16-wide tile) ----
  k_gemm_wmma<HIDDEN, HIDDEN, OUT_DIM, OUT_DIM, 1, OUT_DIM, true>
      <<<gemmBlocks1, 128, 0, stream>>>(B, Wc, bc, out);
}